// ConvIntrinsicLite_26499948216342
// MI455X (gfx1250) — compile-verified
//
#include <hip/hip_runtime.h>

typedef __attribute__((ext_vector_type(2))) float v2f;
typedef __attribute__((ext_vector_type(8))) float v8f;

#define KDIM 40      // N_RADIAL * N_ANGULAR
#define FEAT 16
#define NT   8       // AMT_TEMPLATES
#define ODIM 32      // OUTPUT_DIM

// Block = 256 threads = 8 waves (wave32). Each wave handles a tile of 16 vertices.
// Lane pair (m, m+16) gathers vertex tileBase+m; low lane accumulates features
// {4c,4c+1}, high lane {4c+2,4c+3}  ->  exactly the V_WMMA_F32_16X16X4_F32
// A-matrix VGPR layout for the 16x16 s-tile, chained over 4 K-chunks.
__global__ __launch_bounds__(256)
void conv_intrinsic_lite_kernel(const float* __restrict__ mesh,   // (N,16)
                                const float* __restrict__ bw,     // (N,40,3)
                                const float* __restrict__ coeff,  // (40,40)
                                const float* __restrict__ tw,     // (8,32,16)
                                const float* __restrict__ bias,   // (8,32)
                                const int*   __restrict__ bidx,   // (N,40,3)
                                float*       __restrict__ out,    // (N,32)
                                int N)
{
    __shared__ float lds_tw[NT * ODIM * FEAT];   // 16 KB (of 320 KB/WGP)
    __shared__ float lds_colsum[KDIM];

    // Stage template weights into LDS; fold interp_coeffs over its (r,a) rows.
    for (int i = threadIdx.x; i < NT * ODIM * FEAT; i += blockDim.x)
        lds_tw[i] = tw[i];
    if (threadIdx.x < KDIM) {
        float s = 0.f;
        for (int ra = 0; ra < KDIM; ++ra)
            s += coeff[ra * KDIM + threadIdx.x];
        lds_colsum[threadIdx.x] = s;
    }
    __syncthreads();

    const int lane = threadIdx.x & 31;
    const int wave = threadIdx.x >> 5;
    const int m    = lane & 15;
    const int hi   = lane >> 4;      // 0: features f%4 in {0,1}; 1: {2,3}
    const int fofs = hi * 2;
    const int tileBase = (blockIdx.x * 8 + wave) * 16;

    // ---- Stage 1: barycentric gather + collapsed-prior weighting ----
    // sacc[2c+r] holds s[vertex][4c + r + fofs]   (A-fragment layout)
    float sacc[8] = {0.f, 0.f, 0.f, 0.f, 0.f, 0.f, 0.f, 0.f};
    const int v = tileBase + m;
    if (v < N) {
        const int*   ip = bidx + (long)v * (KDIM * 3);
        const float* wp = bw   + (long)v * (KDIM * 3);
        __builtin_prefetch(ip, 0, 0);   // global_prefetch_b8
        __builtin_prefetch(wp, 0, 0);
        for (int k = 0; k < KDIM; ++k) {
            const float ck = lds_colsum[k];
#pragma unroll
            for (int j = 0; j < 3; ++j) {
                const int   gi = ip[k * 3 + j];
                const float sc = wp[k * 3 + j] * ck;
                const float* row = mesh + (long)gi * FEAT + fofs;  // 8B aligned
#pragma unroll
                for (int c = 0; c < 4; ++c) {
                    const v2f val = *(const v2f*)(row + 4 * c);    // global_load_b64
                    sacc[2 * c + 0] = fmaf(sc, val.x, sacc[2 * c + 0]);
                    sacc[2 * c + 1] = fmaf(sc, val.y, sacc[2 * c + 1]);
                }
            }
        }
    }

    // ---- Stage 2: fold GEMM (16x16 s-tile) x (16x256 W^T) on the matrix pipe ----
    // 16 column tiles; tile ct covers template t = ct>>1, outputs o in
    // [(ct&1)*16, (ct&1)*16+16). ReLU(pre + bias) accumulated over templates.
    v8f accLo = {};   // outputs o = 0..15
    v8f accHi = {};   // outputs o = 16..31
#pragma unroll
    for (int ct = 0; ct < 16; ++ct) {
        const int   t  = ct >> 1;
        const int   oh = ct & 1;
        const int   o  = oh * 16 + m;
        const float bv = bias[t * ODIM + o];
        const float* wbase = &lds_tw[(t * ODIM + o) * FEAT + fofs];

        v8f c = {};
#pragma unroll
        for (int kc = 0; kc < 4; ++kc) {
            v2f a = { sacc[2 * kc], sacc[2 * kc + 1] };
            v2f b = *(const v2f*)(wbase + 4 * kc);                 // ds_load_b64
            c = __builtin_amdgcn_wmma_f32_16x16x4_f32(
                    /*neg_a=*/false, a, /*neg_b=*/false, b,
                    /*c_mod=*/(short)0, c, /*reuse_a=*/false, /*reuse_b=*/false);
        }
#pragma unroll
        for (int i = 0; i < 8; ++i) {
            float x = fmaxf(c[i] + bv, 0.f);
            if (oh) accHi[i] += x; else accLo[i] += x;
        }
    }

    // ---- Store (C/D layout: lanes 0-15 rows M=0..7, lanes 16-31 rows M=8..15) ----
    const int rbase = tileBase + hi * 8;
#pragma unroll
    for (int i = 0; i < 8; ++i) {
        const int vv = rbase + i;
        if (vv < N) {
            out[(long)vv * ODIM + m]      = accLo[i];
            out[(long)vv * ODIM + 16 + m] = accHi[i];
        }
    }
}

extern "C" void kernel_launch(void* const* d_in, const int* in_sizes, int n_in,
                              void* d_out, int out_size, void* d_ws, size_t ws_size,
                              hipStream_t stream) {
    const float* mesh  = (const float*)d_in[0];   // mesh_signal   (N,16)
    const float* bw    = (const float*)d_in[1];   // bary_weights  (N,40,3)
    const float* coeff = (const float*)d_in[2];   // interp_coeffs (40,40)
    const float* tw    = (const float*)d_in[3];   // template_weights (8,32,16)
    const float* bias  = (const float*)d_in[4];   // bias          (8,32)
    const int*   bidx  = (const int*)d_in[5];     // bary_indices  (N,40,3)
    float* out = (float*)d_out;

    const int N = in_sizes[0] / FEAT;
    const int blocks = (N + 127) / 128;           // 128 vertices / block (8 wave-tiles)
    conv_intrinsic_lite_kernel<<<blocks, 256, 0, stream>>>(
        mesh, bw, coeff, tw, bias, bidx, out, N);
}